// GAT_49452253446849
// MI455X (gfx1250) — compile-verified
//
#include <hip/hip_runtime.h>
#include <math.h>

#define NNODES 100000
#define NEDGES 1600000
#define ETOT   (NEDGES + NNODES)

typedef __attribute__((ext_vector_type(2))) float v2f;
typedef __attribute__((ext_vector_type(8))) float v8f;

__device__ __forceinline__ unsigned f2ord(float f) {
    unsigned u = __float_as_uint(f);
    return (u & 0x80000000u) ? ~u : (u | 0x80000000u);
}
__device__ __forceinline__ float ord2f(unsigned u) {
    return (u & 0x80000000u) ? __uint_as_float(u & 0x7fffffffu)
                             : __uint_as_float(~u);
}
__device__ __forceinline__ float lrelu(float x, float s) {
    return x >= 0.0f ? x : s * x;
}

// ---------------------------------------------------------------------------
// GEMM: H[N x DO] = X[N x DI] @ W[DI x DO], fp32 WMMA 16x16x4.
// One wave per 16-row M tile; all DO/16 N-tiles accumulated in registers.
// A layout (32-bit, 16x4): lanes 0-15 row M=lane hold K=0,1; lanes 16-31 hold K=2,3.
// B layout (32-bit, 4x16): lanes 0-15 col N=lane hold K=0,1; lanes 16-31 hold K=2,3.
// C/D (16x16 f32): VGPR r = row r (lanes 0-15) / row r+8 (lanes 16-31), col = lane&15.
// ---------------------------------------------------------------------------
template<int DI, int DO>
__global__ __launch_bounds__(256) void gemm_wmma_kernel(
    const float* __restrict__ X, const float* __restrict__ W,
    float* __restrict__ H)
{
    constexpr int NT = DO / 16;
    const int lane  = threadIdx.x & 31;
    const int mtile = blockIdx.x * (blockDim.x >> 5) + (threadIdx.x >> 5);
    if (mtile * 16 >= NNODES) return;

    const int lrow  = lane & 15;
    const int khalf = (lane >> 4) << 1;          // 0 or 2
    const long rowBase = (long)(mtile * 16 + lrow) * DI;

    v8f zero = {};
    v8f acc[NT];
#pragma unroll
    for (int t = 0; t < NT; ++t) acc[t] = zero;

    for (int k0 = 0; k0 < DI; k0 += 4) {
        const float2 av = *(const float2*)(X + rowBase + k0 + khalf);
        v2f a; a.x = av.x; a.y = av.y;
        const float* Wk = W + (long)(k0 + khalf) * DO;
#pragma unroll
        for (int t = 0; t < NT; ++t) {
            const int col = t * 16 + lrow;
            v2f b; b.x = Wk[col]; b.y = Wk[DO + col];
            acc[t] = __builtin_amdgcn_wmma_f32_16x16x4_f32(
                false, a, false, b, (short)0, acc[t], false, false);
        }
    }

    const int m0 = mtile * 16 + ((lane >> 4) << 3);
#pragma unroll
    for (int t = 0; t < NT; ++t) {
#pragma unroll
        for (int r = 0; r < 8; ++r)
            H[(long)(m0 + r) * DO + t * 16 + lrow] = acc[t][r];
    }
}

// ---------------------------------------------------------------------------
// Per-node attention coefficients: alpha_s[i] = h[i].a_src, alpha_d[i] = h[i].a_dst
// One wave per node, float4 row reads + shuffle reduction.
// ---------------------------------------------------------------------------
__global__ __launch_bounds__(256) void alpha_kernel(
    const float* __restrict__ H,
    const float* __restrict__ avs, const float* __restrict__ avd,
    float* __restrict__ out_s, float* __restrict__ out_d, int dofv)  // dofv = dof/4
{
    const int lane = threadIdx.x & 31;
    const int node = blockIdx.x * (blockDim.x >> 5) + (threadIdx.x >> 5);
    if (node >= NNODES) return;
    const float4* row = (const float4*)(H + (long)node * (dofv * 4));
    const float4* s4  = (const float4*)avs;
    const float4* d4  = (const float4*)avd;
    float ss = 0.0f, sd = 0.0f;
    for (int f = lane; f < dofv; f += 32) {
        const float4 h = row[f];
        const float4 a = s4[f];
        const float4 b = d4[f];
        ss = fmaf(h.x, a.x, fmaf(h.y, a.y, fmaf(h.z, a.z, fmaf(h.w, a.w, ss))));
        sd = fmaf(h.x, b.x, fmaf(h.y, b.y, fmaf(h.z, b.z, fmaf(h.w, b.w, sd))));
    }
#pragma unroll
    for (int o = 16; o > 0; o >>= 1) {
        ss += __shfl_xor(ss, o, 32);
        sd += __shfl_xor(sd, o, 32);
    }
    if (lane == 0) { out_s[node] = ss; out_d[node] = sd; }
}

__global__ void node_init_kernel(unsigned* __restrict__ mbuf, float* __restrict__ sbuf)
{
    const int i = blockIdx.x * blockDim.x + threadIdx.x;
    if (i < NNODES) {
        mbuf[i] = f2ord(-INFINITY);
        sbuf[i] = 0.0f;
    }
}

// sbuf <- 1/sbuf (after segment sums complete)
__global__ void recip_kernel(float* __restrict__ sbuf)
{
    const int i = blockIdx.x * blockDim.x + threadIdx.x;
    if (i < NNODES) sbuf[i] = 1.0f / sbuf[i];
}

__global__ void zero_kernel(float4* __restrict__ p, long n4)
{
    const long i = (long)blockIdx.x * blockDim.x + threadIdx.x;
    if (i < n4) p[i] = make_float4(0.0f, 0.0f, 0.0f, 0.0f);
}

// edge pass 1: logits + segment max (ordered-uint atomicMax)
__global__ void edge_logit_max_kernel(
    const int* __restrict__ src, const int* __restrict__ dst,
    const float* __restrict__ as, const float* __restrict__ ad,
    float* __restrict__ wbuf, unsigned* __restrict__ mbuf)
{
    const int e = blockIdx.x * blockDim.x + threadIdx.x;
    if (e >= ETOT) return;
    const int s = (e < NEDGES) ? src[e] : (e - NEDGES);
    const int d = (e < NEDGES) ? dst[e] : (e - NEDGES);
    const float l = lrelu(as[s] + ad[d], 0.2f);
    wbuf[e] = l;
    atomicMax(mbuf + d, f2ord(l));
}

// edge pass 2: w = exp(logit - max); segment sum
__global__ void edge_expsum_kernel(
    const int* __restrict__ dst, float* __restrict__ wbuf,
    const unsigned* __restrict__ mbuf, float* __restrict__ sbuf)
{
    const int e = blockIdx.x * blockDim.x + threadIdx.x;
    if (e >= ETOT) return;
    const int d = (e < NEDGES) ? dst[e] : (e - NEDGES);
    const float w = __expf(wbuf[e] - ord2f(mbuf[d]));
    wbuf[e] = w;
    atomicAdd(sbuf + d, w);
}

// edge pass 3: acc[dst] += (w * inv_s[dst]) * h[src]
// thread per (edge, float4 chunk): full lane utilization for DOF=64 and 128.
template<int DOF>
__global__ __launch_bounds__(256) void edge_scatter_kernel(
    const int* __restrict__ src, const int* __restrict__ dst,
    const float* __restrict__ wbuf, const float* __restrict__ inv_s,
    const float* __restrict__ H, float* __restrict__ acc)
{
    constexpr int V = DOF / 4;                 // float4 chunks per row (power of 2)
    const long tid = (long)blockIdx.x * blockDim.x + threadIdx.x;
    const long e = tid / V;
    const int  c = (int)(tid & (V - 1));
    if (e >= ETOT) return;
    const int s = (e < NEDGES) ? src[e] : (int)(e - NEDGES);
    const int d = (e < NEDGES) ? dst[e] : (int)(e - NEDGES);
    const float coef = wbuf[e] * inv_s[d];
    const float4 h = *(const float4*)(H + (long)s * DOF + c * 4);
    float* a = acc + (long)d * DOF + c * 4;
    atomicAdd(a + 0, coef * h.x);
    atomicAdd(a + 1, coef * h.y);
    atomicAdd(a + 2, coef * h.z);
    atomicAdd(a + 3, coef * h.w);
}

// bias + leaky_relu(0.25), float4-wide
__global__ void finalize_kernel(
    const float4* __restrict__ accv, const float4* __restrict__ bias,
    float4* __restrict__ out, int maskv, long n4)
{
    const long i = (long)blockIdx.x * blockDim.x + threadIdx.x;
    if (i >= n4) return;
    float4 v = accv[i];
    const float4 b = bias[i & maskv];
    v.x = lrelu(v.x + b.x, 0.25f);
    v.y = lrelu(v.y + b.y, 0.25f);
    v.z = lrelu(v.z + b.z, 0.25f);
    v.w = lrelu(v.w + b.w, 0.25f);
    out[i] = v;
}

// ---------------------------------------------------------------------------

extern "C" void kernel_launch(void* const* d_in, const int* in_sizes, int n_in,
                              void* d_out, int out_size, void* d_ws, size_t ws_size,
                              hipStream_t stream)
{
    const float* X0  = (const float*)d_in[0];
    const int*   adj = (const int*)d_in[1];
    const int*   src = adj;
    const int*   dst = adj + NEDGES;
    const float* W[3]  = {(const float*)d_in[2],  (const float*)d_in[6],  (const float*)d_in[10]};
    const float* As[3] = {(const float*)d_in[3],  (const float*)d_in[7],  (const float*)d_in[11]};
    const float* Ad[3] = {(const float*)d_in[4],  (const float*)d_in[8],  (const float*)d_in[12]};
    const float* Bb[3] = {(const float*)d_in[5],  (const float*)d_in[9],  (const float*)d_in[13]};

    float* ws   = (float*)d_ws;
    float* bufA = ws;                                // H buffer,  N*128
    float* bufB = bufA + (long)NNODES * 128;         // X/ACC,     N*128
    float* asb  = bufB + (long)NNODES * 128;         // N
    float* adb  = asb + NNODES;                      // N
    unsigned* mbuf = (unsigned*)(adb + NNODES);      // N
    float* sbuf = (float*)(mbuf + NNODES);           // N
    float* wbuf = sbuf + NNODES;                     // ETOT

    const int TB = 256;
    const int WPB = TB / 32;
    const dim3 gGemm((NNODES / 16 + WPB - 1) / WPB);
    const dim3 gNodeW((NNODES + WPB - 1) / WPB);
    const dim3 gNode((NNODES + TB - 1) / TB);
    const dim3 gEdge((ETOT + TB - 1) / TB);
    const dim3 gScat128((unsigned)(((long)ETOT * 32 + TB - 1) / TB));
    const dim3 gScat64((unsigned)(((long)ETOT * 16 + TB - 1) / TB));
    const long n128_4 = (long)NNODES * 32;           // N*128 in float4
    const long n64_4  = (long)NNODES * 16;           // N*64  in float4
    const dim3 g128((unsigned)((n128_4 + TB - 1) / TB));
    const dim3 g64((unsigned)((n64_4 + TB - 1) / TB));

    // ---- Layer 1: 512 -> 128 ----
    gemm_wmma_kernel<512,128><<<gGemm, TB, 0, stream>>>(X0, W[0], bufA);
    alpha_kernel<<<gNodeW, TB, 0, stream>>>(bufA, As[0], Ad[0], asb, adb, 32);
    node_init_kernel<<<gNode, TB, 0, stream>>>(mbuf, sbuf);
    zero_kernel<<<g128, TB, 0, stream>>>((float4*)bufB, n128_4);
    edge_logit_max_kernel<<<gEdge, TB, 0, stream>>>(src, dst, asb, adb, wbuf, mbuf);
    edge_expsum_kernel<<<gEdge, TB, 0, stream>>>(dst, wbuf, mbuf, sbuf);
    recip_kernel<<<gNode, TB, 0, stream>>>(sbuf);
    edge_scatter_kernel<128><<<gScat128, TB, 0, stream>>>(src, dst, wbuf, sbuf, bufA, bufB);
    finalize_kernel<<<g128, TB, 0, stream>>>((const float4*)bufB, (const float4*)Bb[0],
                                             (float4*)bufB, 31, n128_4);

    // ---- Layer 2: 128 -> 128 ----
    gemm_wmma_kernel<128,128><<<gGemm, TB, 0, stream>>>(bufB, W[1], bufA);
    alpha_kernel<<<gNodeW, TB, 0, stream>>>(bufA, As[1], Ad[1], asb, adb, 32);
    node_init_kernel<<<gNode, TB, 0, stream>>>(mbuf, sbuf);
    zero_kernel<<<g128, TB, 0, stream>>>((float4*)bufB, n128_4);  // bufB dead after gemm
    edge_logit_max_kernel<<<gEdge, TB, 0, stream>>>(src, dst, asb, adb, wbuf, mbuf);
    edge_expsum_kernel<<<gEdge, TB, 0, stream>>>(dst, wbuf, mbuf, sbuf);
    recip_kernel<<<gNode, TB, 0, stream>>>(sbuf);
    edge_scatter_kernel<128><<<gScat128, TB, 0, stream>>>(src, dst, wbuf, sbuf, bufA, bufB);
    finalize_kernel<<<g128, TB, 0, stream>>>((const float4*)bufB, (const float4*)Bb[1],
                                             (float4*)bufB, 31, n128_4);

    // ---- Layer 3: 128 -> 64 ----
    gemm_wmma_kernel<128,64><<<gGemm, TB, 0, stream>>>(bufB, W[2], bufA);
    alpha_kernel<<<gNodeW, TB, 0, stream>>>(bufA, As[2], Ad[2], asb, adb, 16);
    node_init_kernel<<<gNode, TB, 0, stream>>>(mbuf, sbuf);
    zero_kernel<<<g64, TB, 0, stream>>>((float4*)bufB, n64_4);
    edge_logit_max_kernel<<<gEdge, TB, 0, stream>>>(src, dst, asb, adb, wbuf, mbuf);
    edge_expsum_kernel<<<gEdge, TB, 0, stream>>>(dst, wbuf, mbuf, sbuf);
    recip_kernel<<<gNode, TB, 0, stream>>>(sbuf);
    edge_scatter_kernel<64><<<gScat64, TB, 0, stream>>>(src, dst, wbuf, sbuf, bufA, bufB);
    finalize_kernel<<<g64, TB, 0, stream>>>((const float4*)bufB, (const float4*)Bb[2],
                                            (float4*)d_out, 15, n64_4);
}